// SpeciesDetector_24515673325990
// MI455X (gfx1250) — compile-verified
//
#include <hip/hip_runtime.h>
#include <hip/hip_bf16.h>
#include <math.h>

// Problem constants (from reference): N=64, M=8, G=64, D=4, TAU=5
#define PN 64
#define PM 8
#define PG 64
#define GG (PG*PG)          // 4096 grid points
#define P_TILE 16           // grid points per block
#define N_TILE 4            // n values per block
#define TAU_INV 0.2f

typedef __attribute__((ext_vector_type(2))) float v2f;
typedef __attribute__((ext_vector_type(8))) float v8f;

// --- 4x4 determinant via 2x2 sub-determinants (row-major a[16]) ---
__device__ __forceinline__ float det4(const float* a) {
    float s0 = a[0]*a[5]-a[4]*a[1];
    float s1 = a[0]*a[6]-a[4]*a[2];
    float s2 = a[0]*a[7]-a[4]*a[3];
    float s3 = a[1]*a[6]-a[5]*a[2];
    float s4 = a[1]*a[7]-a[5]*a[3];
    float s5 = a[2]*a[7]-a[6]*a[3];
    float c5 = a[10]*a[15]-a[14]*a[11];
    float c4 = a[9]*a[15]-a[13]*a[11];
    float c3 = a[9]*a[14]-a[13]*a[10];
    float c2 = a[8]*a[15]-a[12]*a[11];
    float c1 = a[8]*a[14]-a[12]*a[10];
    float c0 = a[8]*a[13]-a[12]*a[9];
    return s0*c5 - s1*c4 + s2*c3 + s3*c2 - s4*c1 + s5*c0;
}

// --- 4x4 inverse via adjugate; returns determinant ---
__device__ __forceinline__ float inv4(const float* a, float* inv) {
    float s0 = a[0]*a[5]-a[4]*a[1];
    float s1 = a[0]*a[6]-a[4]*a[2];
    float s2 = a[0]*a[7]-a[4]*a[3];
    float s3 = a[1]*a[6]-a[5]*a[2];
    float s4 = a[1]*a[7]-a[5]*a[3];
    float s5 = a[2]*a[7]-a[6]*a[3];
    float c5 = a[10]*a[15]-a[14]*a[11];
    float c4 = a[9]*a[15]-a[13]*a[11];
    float c3 = a[9]*a[14]-a[13]*a[10];
    float c2 = a[8]*a[15]-a[12]*a[11];
    float c1 = a[8]*a[14]-a[12]*a[10];
    float c0 = a[8]*a[13]-a[12]*a[9];
    float det = s0*c5 - s1*c4 + s2*c3 + s3*c2 - s4*c1 + s5*c0;
    float id  = 1.0f / det;
    inv[0]  = ( a[5]*c5 - a[6]*c4 + a[7]*c3) * id;
    inv[1]  = (-a[1]*c5 + a[2]*c4 - a[3]*c3) * id;
    inv[2]  = ( a[13]*s5 - a[14]*s4 + a[15]*s3) * id;
    inv[3]  = (-a[9]*s5 + a[10]*s4 - a[11]*s3) * id;
    inv[4]  = (-a[4]*c5 + a[6]*c2 - a[7]*c1) * id;
    inv[5]  = ( a[0]*c5 - a[2]*c2 + a[3]*c1) * id;
    inv[6]  = (-a[12]*s5 + a[14]*s2 - a[15]*s1) * id;
    inv[7]  = ( a[8]*s5 - a[10]*s2 + a[11]*s1) * id;
    inv[8]  = ( a[4]*c4 - a[5]*c2 + a[7]*c0) * id;
    inv[9]  = (-a[0]*c4 + a[1]*c2 - a[3]*c0) * id;
    inv[10] = ( a[12]*s4 - a[13]*s2 + a[15]*s0) * id;
    inv[11] = (-a[8]*s4 + a[9]*s2 - a[11]*s0) * id;
    inv[12] = (-a[4]*c3 + a[5]*c1 - a[6]*c0) * id;
    inv[13] = ( a[0]*c3 - a[1]*c1 + a[2]*c0) * id;
    inv[14] = (-a[12]*s3 + a[13]*s1 - a[14]*s0) * id;
    inv[15] = ( a[8]*s3 - a[9]*s1 + a[10]*s0) * id;
    return det;
}

__global__ __launch_bounds__(256)
void species_kl_kernel(const float* __restrict__ mu_a,
                       const float* __restrict__ sigma_a,
                       const float* __restrict__ omega_a,
                       const float* __restrict__ mu_b,
                       const float* __restrict__ sigma_b,
                       const float* __restrict__ omega_b,
                       const float* __restrict__ chi,
                       float* __restrict__ out) {
    // Block covers: P_TILE consecutive grid points, N_TILE n values, all M=8 m values.
    const int pBase = blockIdx.x * P_TILE;   // gridDim.x = GG/P_TILE = 256
    const int nBase = blockIdx.y * N_TILE;   // gridDim.y = PN/N_TILE = 16
    const int t = threadIdx.x;               // 256 threads = 8 wave32

    __shared__ float sOa  [N_TILE * P_TILE * 16];   // 1024
    __shared__ float sSa  [N_TILE * P_TILE * 16];   // 1024
    __shared__ float sObI [PM     * P_TILE * 16];   // 2048 (omega_b, inverted in place)
    __shared__ float sSb  [PM     * P_TILE * 16];   // 2048
    __shared__ float sMuA [N_TILE * P_TILE * 4];    // 256
    __shared__ float sMuB [PM     * P_TILE * 4];    // 512
    __shared__ float sChi [N_TILE * P_TILE];        // 64
    __shared__ float sLdA [N_TILE * P_TILE];        // 64
    __shared__ float sT   [P_TILE * N_TILE * PM * 16]; // 8192 (per-pair transport matrices)

    // ---------------- Stage 0: cooperative staging (float4, coalesced) -------------
    {   // omega_a, sigma_a: 1024 floats each = 256 float4; 1 per thread.
        int nLoc = t >> 6;          // 64 float4 per n-span of 256 floats
        int rem  = t & 63;
        size_t gbase = ((size_t)(nBase + nLoc) * GG + pBase) * 16;
        ((float4*)sOa)[t] = ((const float4*)(omega_a + gbase))[rem];
        ((float4*)sSa)[t] = ((const float4*)(sigma_a + gbase))[rem];
    }
    #pragma unroll
    for (int r = 0; r < 2; ++r) {   // omega_b, sigma_b: 2048 floats = 512 float4; 2 per thread
        int ft  = t + r * 256;
        int m   = ft >> 6;
        int rem = ft & 63;
        size_t gbase = ((size_t)m * GG + pBase) * 16;
        ((float4*)sObI)[ft] = ((const float4*)(omega_b + gbase))[rem];
        ((float4*)sSb )[ft] = ((const float4*)(sigma_b + gbase))[rem];
    }
    if (t < 64) {                   // mu_a: 256 floats = 64 float4
        int nLoc = t >> 4, rem = t & 15;
        size_t gbase = ((size_t)(nBase + nLoc) * GG + pBase) * 4;
        ((float4*)sMuA)[t] = ((const float4*)(mu_a + gbase))[rem];
    } else if (t < 192) {           // mu_b: 512 floats = 128 float4
        int ft = t - 64;
        int m = ft >> 4, rem = ft & 15;
        size_t gbase = ((size_t)m * GG + pBase) * 4;
        ((float4*)sMuB)[ft] = ((const float4*)(mu_b + gbase))[rem];
    } else {                        // chi: 64 floats
        int ft = t - 192;
        int nLoc = ft >> 4, pLoc = ft & 15;
        sChi[ft] = chi[(size_t)(nBase + nLoc) * GG + pBase + pLoc];
    }
    __syncthreads();

    // ---------------- Stage 1: precompute log|Sa| and invert omega_b ---------------
    if (t < 64) {                   // 4n x 16p slogdet(sigma_a)
        sLdA[t] = logf(det4(sSa + t * 16));
    } else if (t >= 128) {          // 8m x 16p: omega_b^-1 in place
        int tt = t - 128;
        float* A = sObI + tt * 16;
        float inv[16];
        inv4(A, inv);
        #pragma unroll
        for (int e = 0; e < 16; ++e) A[e] = inv[e];
    }
    __syncthreads();

    // ---------------- Stage 2: transport via V_WMMA_F32_16X16X4_F32 ----------------
    // One WMMA = 16 cross 4x4 products: block(i,j) of C = Oa[nBase+i] @ ObInv[mTile*4+j].
    // 32 WMMAs per block (16 p x 2 m-tiles), 4 per wave. EXEC all-ones (no divergence).
    {
        const int wave = t >> 5;
        const int lane = t & 31;
        const int row  = lane & 15;     // A row (M) / B column (N)
        const int hi   = lane >> 4;     // K half: lanes 16-31 carry K=2,3 (ISA 16x4 f32 layout)
        #pragma unroll
        for (int it = 0; it < 4; ++it) {
            const int wid  = wave * 4 + it;     // 0..31
            const int pLoc = wid >> 1;
            const int mT   = wid & 1;
            // A[M][K] = Oa[nBase + M/4][M%4][K]
            const int nLoc = row >> 2, i = row & 3;
            const float* Ab = sOa + ((nLoc * P_TILE + pLoc) * 16) + i * 4;
            v2f a; a.x = Ab[hi ? 2 : 0]; a.y = Ab[hi ? 3 : 1];
            // B[K][N] = ObInv[mT*4 + N/4][K][N%4]
            const int mLoc = mT * 4 + (row >> 2), j = row & 3;
            const float* Bb = sObI + ((mLoc * P_TILE + pLoc) * 16) + j;
            v2f b; b.x = Bb[(hi ? 2 : 0) * 4]; b.y = Bb[(hi ? 3 : 1) * 4];
            v8f c = {0.f, 0.f, 0.f, 0.f, 0.f, 0.f, 0.f, 0.f};
            c = __builtin_amdgcn_wmma_f32_16x16x4_f32(
                    /*neg_a=*/false, a, /*neg_b=*/false, b,
                    /*c_mod=*/(short)0, c, /*reuse_a=*/false, /*reuse_b=*/false);
            // D layout: VGPR v -> row v (lanes 0-15) / row v+8 (lanes 16-31), col = lane&15
            #pragma unroll
            for (int v = 0; v < 8; ++v) {
                int rM = v + hi * 8;
                int nL = rM >> 2, ii = rM & 3;
                int mL = mT * 4 + (row >> 2), jj = row & 3;
                sT[(((pLoc * N_TILE) + nL) * PM + mL) * 16 + ii * 4 + jj] = c[v];
            }
        }
    }
    __syncthreads();

    // ---------------- Stage 3: per-pair KL (VALU), coalesced store -----------------
    #pragma unroll
    for (int half = 0; half < 2; ++half) {
        const int q    = t + half * 256;          // 512 pairs per block
        const int pLoc = q & 15;                  // fastest -> contiguous output runs
        const int nm   = q >> 4;
        const int nLoc = nm >> 3;
        const int m    = nm & 7;

        const float* T  = sT  + (((pLoc * N_TILE) + nLoc) * PM + m) * 16;
        const float* Sb = sSb + (m    * P_TILE + pLoc) * 16;
        const float* Sa = sSa + (nLoc * P_TILE + pLoc) * 16;
        const float* mb = sMuB + (m    * P_TILE + pLoc) * 4;
        const float* ma = sMuA + (nLoc * P_TILE + pLoc) * 4;

        // St = T * Sb * T^T
        float X[16], St[16], StI[16];
        #pragma unroll
        for (int i = 0; i < 4; ++i)
            #pragma unroll
            for (int k = 0; k < 4; ++k) {
                float s = 0.f;
                #pragma unroll
                for (int jj = 0; jj < 4; ++jj) s += T[i*4+jj] * Sb[jj*4+k];
                X[i*4+k] = s;
            }
        #pragma unroll
        for (int i = 0; i < 4; ++i)
            #pragma unroll
            for (int l = 0; l < 4; ++l) {
                float s = 0.f;
                #pragma unroll
                for (int k = 0; k < 4; ++k) s += X[i*4+k] * T[l*4+k];
                St[i*4+l] = s;
            }
        float det  = inv4(St, StI);
        float ld_b = logf(det);

        float diff[4];
        #pragma unroll
        for (int i = 0; i < 4; ++i) {
            float s = 0.f;
            #pragma unroll
            for (int jj = 0; jj < 4; ++jj) s += T[i*4+jj] * mb[jj];
            diff[i] = ma[i] - s;
        }
        float tr = 0.f, mahal = 0.f;
        #pragma unroll
        for (int i = 0; i < 4; ++i) {
            float s = 0.f;
            #pragma unroll
            for (int jj = 0; jj < 4; ++jj) {
                tr += StI[i*4+jj] * Sa[jj*4+i];
                s  += StI[i*4+jj] * diff[jj];
            }
            mahal += diff[i] * s;
        }
        float kl = 0.5f * (tr + mahal - 4.0f + ld_b - sLdA[nLoc * P_TILE + pLoc]);
        float Sv = sChi[nLoc * P_TILE + pLoc] / (1.0f + expf(kl * TAU_INV));

        const int n = nBase + nLoc;
        out[(size_t)(n * PM + m) * GG + pBase + pLoc] = Sv;
    }
}

extern "C" void kernel_launch(void* const* d_in, const int* in_sizes, int n_in,
                              void* d_out, int out_size, void* d_ws, size_t ws_size,
                              hipStream_t stream) {
    const float* mu_a    = (const float*)d_in[0];
    const float* sigma_a = (const float*)d_in[1];
    const float* omega_a = (const float*)d_in[2];
    const float* mu_b    = (const float*)d_in[3];
    const float* sigma_b = (const float*)d_in[4];
    const float* omega_b = (const float*)d_in[5];
    const float* chi     = (const float*)d_in[6];
    float* out = (float*)d_out;

    dim3 grid(GG / P_TILE, PN / N_TILE);   // (256, 16)
    dim3 block(256);                       // 8 wave32
    hipLaunchKernelGGL(species_kl_kernel, grid, block, 0, stream,
                       mu_a, sigma_a, omega_a, mu_b, sigma_b, omega_b, chi, out);
}